// Cafe_RNN_24945170055817
// MI455X (gfx1250) — compile-verified
//
#include <hip/hip_runtime.h>
#include <cstdint>

// Grid: B=1, D=128. x: (D,D,D,5) [state, euler*3, field]; out: (D,D,D,8);
// result: (D,D,D,5) = [state1, euler1*3, field1].
#define D 128
#define TI 8
#define TJ 8
#define TK 8
#define HI (TI + 2)
#define HJ (TJ + 2)
#define HK (TK + 2)
#define HN (HI * HJ * HK)        // 1000 halo voxels
#define NTHREADS 512

// CDNA5 async global->LDS copy (GV mode, saddr=off). lds_off is the byte
// offset within the workgroup's LDS allocation (single __shared__ array
// => allocated at offset 0). Tracked by ASYNCcnt.
__device__ __forceinline__ void async_g2l_b32(uint32_t lds_off, const float* g) {
    asm volatile("global_load_async_to_lds_b32 %0, %1, off"
                 :
                 : "v"(lds_off), "v"(g)
                 : "memory");
}

__device__ __forceinline__ void wait_asynccnt0() {
    asm volatile("s_wait_asynccnt 0x0" ::: "memory");
}

__global__ __launch_bounds__(NTHREADS)
void Cafe_RNN_kernel(const float* __restrict__ x,
                     const float* __restrict__ out,
                     float* __restrict__ res) {
    // Halo euler tile, padded to float4 per voxel (16 B) so every neighbor
    // read is one naturally-aligned ds_load_b128. 16000 bytes at LDS offset 0.
    __shared__ float4 sE[HN];

    const int tid = threadIdx.x;
    const int i0 = blockIdx.z * TI;
    const int j0 = blockIdx.y * TJ;
    const int k0 = blockIdx.x * TK;

    // ---- Stage (TI+2)x(TJ+2)x(TK+2) euler halo into LDS via async loads ----
    for (int t = tid; t < HN; t += NTHREADS) {
        const int hk = t % HK;
        const int hj = (t / HK) % HJ;
        const int hi = t / (HJ * HK);
        const int gi = i0 + hi - 1;
        const int gj = j0 + hj - 1;
        const int gk = k0 + hk - 1;
        const uint32_t off = (uint32_t)t * 16u;   // float4 slot per halo voxel
        const bool inb = (gi >= 0) & (gi < D) & (gj >= 0) & (gj < D) &
                         (gk >= 0) & (gk < D);
        if (inb) {
            const float* g = x + (size_t)((gi * D + gj) * D + gk) * 5 + 1;
            async_g2l_b32(off,      g);       // euler.x
            async_g2l_b32(off + 4,  g + 1);   // euler.y
            async_g2l_b32(off + 8,  g + 2);   // euler.z  (.w slot unused)
        } else {
            // zero padding (reference pads euler with zeros)
            sE[t] = make_float4(0.0f, 0.0f, 0.0f, 0.0f);
        }
    }
    wait_asynccnt0();   // async writes landed in LDS
    __syncthreads();    // visible to all waves (also covers the DS zero-fills)

    // ---- Per-voxel update (one thread per voxel of the 8x8x8 tile) ----
    const int li = tid >> 6;
    const int lj = (tid >> 3) & 7;
    const int lk = tid & 7;
    const int gi = i0 + li;
    const int gj = j0 + lj;
    const int gk = k0 + lk;
    const size_t idx = (size_t)((gi * D + gj) * D + gk);

    // out record: 8 floats, 32B-aligned -> two b128 loads
    const float4* o4 = (const float4*)(out + idx * 8);
    const float4 oa = o4[0];   // logits 0..3
    const float4 ob = o4[1];   // rot1.xyz, field raw

    const int state0 = (int)x[idx * 5];   // only the state float from x

    // argmax over 4 logits, first-max tie-break (matches jnp.argmax)
    int s1 = 0;
    float best = oa.x;
    if (oa.y > best) { best = oa.y; s1 = 1; }
    if (oa.z > best) { best = oa.z; s1 = 2; }
    if (oa.w > best) { best = oa.w; s1 = 3; }
    if (state0 == 0) s1 = 0;

    float field1 = ob.w;
    if (s1 <= 1)      field1 = -1.0f;
    else if (s1 == 2) field1 = fminf(fmaxf(field1, 0.0f), 0.92f);
    else              field1 = 1.0f;

    // own euler from LDS (interior of halo tile), one ds_load_b128
    const int cc = ((li + 1) * HJ + (lj + 1)) * HK + (lk + 1);
    const float4 ec = sE[cc];
    float ex = ec.x, ey = ec.y, ez = ec.z;
    if (s1 <= 1) { ex = -1.0f; ey = -1.0f; ez = -1.0f; }

    // solidify: 26-neighbor argmin of squared distance to rot1
    if ((state0 <= 1) && (s1 > 1)) {
        const float rx = ob.x, ry = ob.y, rz = ob.z;
        float dmin = INFINITY;
        float cx = 0.0f, cy = 0.0f, cz = 0.0f;
        #pragma unroll
        for (int di = -1; di <= 1; ++di) {
            #pragma unroll
            for (int dj = -1; dj <= 1; ++dj) {
                #pragma unroll
                for (int dk = -1; dk <= 1; ++dk) {
                    if (di == 0 && dj == 0 && dk == 0) continue;
                    const int nb =
                        ((li + 1 + di) * HJ + (lj + 1 + dj)) * HK + (lk + 1 + dk);
                    const float4 n = sE[nb];          // single ds_load_b128
                    const float dx = n.x - rx;
                    const float dy = n.y - ry;
                    const float dz = n.z - rz;
                    const float d = dx * dx + dy * dy + dz * dz;
                    if (d < dmin) { dmin = d; cx = n.x; cy = n.y; cz = n.z; }
                }
            }
        }
        ex = cx; ey = cy; ez = cz;
    }

    float* r = res + idx * 5;
    r[0] = (float)s1;
    r[1] = ex;
    r[2] = ey;
    r[3] = ez;
    r[4] = field1;
}

extern "C" void kernel_launch(void* const* d_in, const int* in_sizes, int n_in,
                              void* d_out, int out_size, void* d_ws, size_t ws_size,
                              hipStream_t stream) {
    (void)in_sizes; (void)n_in; (void)out_size; (void)d_ws; (void)ws_size;
    const float* x   = (const float*)d_in[0];   // (1,128,128,128,5) f32
    const float* out = (const float*)d_in[1];   // (1,128,128,128,8) f32
    float* res       = (float*)d_out;           // (1,128,128,128,5) f32

    dim3 grid(D / TK, D / TJ, D / TI);          // 16 x 16 x 16 blocks
    Cafe_RNN_kernel<<<grid, NTHREADS, 0, stream>>>(x, out, res);
}